// GTLayer_83554293776837
// MI455X (gfx1250) — compile-verified
//
#include <hip/hip_runtime.h>
#include <hip/hip_bf16.h>
#include <math.h>

typedef __attribute__((ext_vector_type(2))) float v2f;
typedef __attribute__((ext_vector_type(8))) float v8f;

#define HDIM 64
#define NHEAD 8
#define DHEAD 8

#if defined(__HIP_DEVICE_COMPILE__) && \
    __has_builtin(__builtin_amdgcn_global_load_async_to_lds_b128)
#define ASYNC_STAGE 1
#else
#define ASYNC_STAGE 0
#endif

// ---------------------------------------------------------------------------
// Stage a 64x64 f32 weight matrix (1024 float4) from global into LDS.
// Async global->LDS (ASYNCcnt) when available; plain b128 copies otherwise.
// Caller must wait (stage_wait) + __syncthreads() before reading LDS.
// ---------------------------------------------------------------------------
__device__ __forceinline__ void stage_weight(const float* __restrict__ Wsrc,
                                             float* dstLds, int tid) {
#if ASYNC_STAGE
    typedef __attribute__((ext_vector_type(4))) int v4i;
    typedef __attribute__((address_space(1))) v4i gv4i;
    typedef __attribute__((address_space(3))) v4i lv4i;
    gv4i* g = (gv4i*)Wsrc;
    lv4i* l = (lv4i*)dstLds;
    for (int i = tid; i < 1024; i += 256)
        __builtin_amdgcn_global_load_async_to_lds_b128(g + i, l + i, 0, 0);
#else
    const float4* s = (const float4*)Wsrc;
    float4* d = (float4*)dstLds;
    for (int i = tid; i < 1024; i += 256) d[i] = s[i];
#endif
}

__device__ __forceinline__ void stage_wait() {
#if ASYNC_STAGE
#if __has_builtin(__builtin_amdgcn_s_wait_asynccnt)
    __builtin_amdgcn_s_wait_asynccnt(0);
#else
    asm volatile("s_wait_asynccnt 0x0" ::: "memory");
#endif
#endif
}

// ---------------------------------------------------------------------------
// ordered-uint mapping for float atomic max (monotonic bijection)
// ---------------------------------------------------------------------------
__device__ __forceinline__ unsigned f2ord(float f) {
    unsigned u = __float_as_uint(f);
    return (u & 0x80000000u) ? ~u : (u | 0x80000000u);
}
__device__ __forceinline__ float ord2f(unsigned u) {
    return (u & 0x80000000u) ? __uint_as_float(u & 0x7FFFFFFFu)
                             : __uint_as_float(~u);
}
#define ORD_NEG_INF 0x007FFFFFu  // f2ord(-inf)

// ---------------------------------------------------------------------------
// Kernel 1: AtomEncoder  h[n,c] = sum_i atom_emb[X[n,i]+off[i], c]
// ---------------------------------------------------------------------------
__global__ void atom_encoder_kernel(const int* __restrict__ X,
                                    const float* __restrict__ atom_emb,
                                    float* __restrict__ h, int N) {
    int idx = blockIdx.x * blockDim.x + threadIdx.x;
    if (idx >= N * HDIM) return;
    int n = idx >> 6;
    int c = idx & 63;
    const int offs[9] = {0, 119, 124, 136, 148, 158, 164, 170, 172};
    const int* xr = X + (size_t)n * 9;
    float acc = 0.f;
#pragma unroll
    for (int i = 0; i < 9; ++i) {
        int r = xr[i] + offs[i];
        acc += atom_emb[(size_t)r * HDIM + c];
    }
    h[idx] = acc;
}

// ---------------------------------------------------------------------------
// Kernel 2: fused QKV projection with V_WMMA_F32_16X16X4_F32.
//   256 threads = 8 waves; each wave owns a 16-row panel; block covers 128 rows.
//   Wq/Wk/Wv staged in LDS (48 KB) via async global->LDS. Outputs stored
//   HEAD-MAJOR: q[n][h*8+d] = (h @ Wq + bq)[n][d*8+h] * scale (so edge kernels
//   can use contiguous 32B float4 loads per (edge, head)).
// ---------------------------------------------------------------------------
__global__ __launch_bounds__(256) void qkv_wmma_kernel(
    const float* __restrict__ h,
    const float* __restrict__ Wq, const float* __restrict__ bq,
    const float* __restrict__ Wk, const float* __restrict__ bk,
    const float* __restrict__ Wv, const float* __restrict__ bv,
    float* __restrict__ q, float* __restrict__ k, float* __restrict__ v,
    int N) {
    __shared__ float sW[3 * HDIM * HDIM];
    stage_weight(Wq, sW, threadIdx.x);
    stage_weight(Wk, sW + 4096, threadIdx.x);
    stage_weight(Wv, sW + 8192, threadIdx.x);
    stage_wait();
    __syncthreads();

    const int lane = threadIdx.x & 31;
    const int wave = threadIdx.x >> 5;
    const int rowBase = blockIdx.x * 128 + wave * 16;
    if (rowBase >= N) return;  // uniform per wave

    const int mrow  = lane & 15;  // A: M index; B/C/D: N index
    const int half  = lane >> 4;  // 0 or 1
    const int khalf = half * 2;   // A/B K sub-offset within a K=4 chunk

    int gr = rowBase + mrow;
    if (gr >= N) gr = N - 1;  // clamp: EXEC stays all-ones for WMMA

    // A fragments: full K=64 panel for this wave's 16 rows (16 chunks of K=4)
    v2f afrag[16];
#pragma unroll
    for (int kc = 0; kc < 16; ++kc)
        afrag[kc] = *(const v2f*)(h + (size_t)gr * HDIM + kc * 4 + khalf);

    const float scaling = 0.35355339059327379f;  // DH^-0.5
    const float* biases[3] = {bq, bk, bv};
    float* outs[3] = {q, k, v};

    for (int w = 0; w < 3; ++w) {
        const float* Wl = sW + w * 4096;
        const float* bias = biases[w];
        float* Out = outs[w];
        const float sc = (w == 0) ? scaling : 1.0f;
#pragma unroll
        for (int nt = 0; nt < 4; ++nt) {
            const int n0 = nt * 16;
            const int cc = n0 + mrow;                  // logical column
            const int pcol = (cc & 7) * 8 + (cc >> 3); // head-major column
            const float bb = bias[cc];
            v8f c;
#pragma unroll
            for (int j = 0; j < 8; ++j) c[j] = bb;
#pragma unroll
            for (int kc = 0; kc < 16; ++kc) {
                const int kb = kc * 4 + khalf;
                v2f b;
                b.x = Wl[(kb + 0) * HDIM + cc];
                b.y = Wl[(kb + 1) * HDIM + cc];
                c = __builtin_amdgcn_wmma_f32_16x16x4_f32(
                        false, afrag[kc], false, b, (short)0, c, false, false);
            }
#pragma unroll
            for (int j = 0; j < 8; ++j) {
                const int m = half * 8 + j;  // C/D row layout
                const int grr = rowBase + m;
                if (grr < N)
                    Out[(size_t)grr * HDIM + pcol] = c[j] * sc;
            }
        }
    }
}

// ---------------------------------------------------------------------------
// Kernel 3: init accumulators (acc aliases h), segment max / sum state
// ---------------------------------------------------------------------------
__global__ void init_kernel(float* __restrict__ acc,
                            unsigned* __restrict__ m_ord,
                            float* __restrict__ ssum, int N) {
    int idx = blockIdx.x * blockDim.x + threadIdx.x;
    if (idx < N * HDIM) acc[idx] = 0.f;
    if (idx < N * NHEAD) {
        m_ord[idx] = ORD_NEG_INF;
        ssum[idx] = 0.f;
    }
}

// ---------------------------------------------------------------------------
// Kernel 4: SDDMM + atomic segment-max   (one thread per (edge, head))
//   q/k are head-major: the per-head 8-vector is 32B contiguous -> float4 x2.
// ---------------------------------------------------------------------------
__global__ void sddmm_max_kernel(const float* __restrict__ q,
                                 const float* __restrict__ k,
                                 const int* __restrict__ row,
                                 const int* __restrict__ col,
                                 float* __restrict__ score,
                                 unsigned* __restrict__ m_ord, int E) {
    int idx = blockIdx.x * blockDim.x + threadIdx.x;
    if (idx >= E * NHEAD) return;
    const int e = idx >> 3;
    const int hh = idx & 7;
    const int r = row[e];
    const int c = col[e];
    const float4* qp = (const float4*)(q + (size_t)r * HDIM + hh * DHEAD);
    const float4* kp = (const float4*)(k + (size_t)c * HDIM + hh * DHEAD);
    const float4 a0 = qp[0], a1 = qp[1];
    const float4 b0 = kp[0], b1 = kp[1];
    float s = a0.x * b0.x + a0.y * b0.y + a0.z * b0.z + a0.w * b0.w +
              a1.x * b1.x + a1.y * b1.y + a1.z * b1.z + a1.w * b1.w;
    score[idx] = s;
    atomicMax(&m_ord[r * NHEAD + hh], f2ord(s));
}

// ---------------------------------------------------------------------------
// Kernel 5 (fused): p = exp(score - m[row]); ssum += p; acc += p * v[col]
//   Normalization by ssum is deferred to the output projection.
// ---------------------------------------------------------------------------
__global__ void exp_spmm_kernel(const float* __restrict__ score,
                                const float* __restrict__ v,
                                const int* __restrict__ row,
                                const int* __restrict__ col,
                                const unsigned* __restrict__ m_ord,
                                float* __restrict__ ssum,
                                float* __restrict__ acc, int E) {
    int idx = blockIdx.x * blockDim.x + threadIdx.x;
    if (idx >= E * NHEAD) return;
    const int e = idx >> 3;
    const int hh = idx & 7;
    const int r = row[e];
    const int c = col[e];
    const float mval = ord2f(m_ord[r * NHEAD + hh]);
    const float p = expf(score[idx] - mval);
    atomicAdd(&ssum[r * NHEAD + hh], p);
    const float4* vp = (const float4*)(v + (size_t)c * HDIM + hh * DHEAD);
    const float4 v0 = vp[0], v1 = vp[1];
    float* ap = acc + (size_t)r * HDIM + hh * DHEAD;
    atomicAdd(ap + 0, p * v0.x);
    atomicAdd(ap + 1, p * v0.y);
    atomicAdd(ap + 2, p * v0.z);
    atomicAdd(ap + 3, p * v0.w);
    atomicAdd(ap + 4, p * v1.x);
    atomicAdd(ap + 5, p * v1.y);
    atomicAdd(ap + 6, p * v1.z);
    atomicAdd(ap + 7, p * v1.w);
}

// ---------------------------------------------------------------------------
// Kernel 6: output projection  out = (acc / ssum) @ Wo + bo  (WMMA f32)
//   acc is head-major and unnormalized; normalize + un-permute at A-load.
// ---------------------------------------------------------------------------
__global__ __launch_bounds__(256) void out_proj_wmma_kernel(
    const float* __restrict__ acc, const float* __restrict__ ssum,
    const float* __restrict__ Wo, const float* __restrict__ bo,
    float* __restrict__ out, int N) {
    __shared__ float sW[HDIM * HDIM];
    stage_weight(Wo, sW, threadIdx.x);
    stage_wait();
    __syncthreads();

    const int lane = threadIdx.x & 31;
    const int wave = threadIdx.x >> 5;
    const int rowBase = blockIdx.x * 128 + wave * 16;
    if (rowBase >= N) return;

    const int mrow = lane & 15;
    const int half = lane >> 4;
    const int khalf = half * 2;

    int gr = rowBase + mrow;
    if (gr >= N) gr = N - 1;

    // per-(node, head) inverse softmax denominators (0 for empty rows)
    float inv[NHEAD];
#pragma unroll
    for (int hh = 0; hh < NHEAD; ++hh) {
        const float sv = ssum[(size_t)gr * NHEAD + hh];
        inv[hh] = (sv > 0.f) ? (1.0f / sv) : 0.f;
    }

    // A fragments: logical column c = d*8+h lives at head-major h*8+d
    v2f afrag[16];
#pragma unroll
    for (int kc = 0; kc < 16; ++kc) {
#pragma unroll
        for (int j = 0; j < 2; ++j) {
            const int c = kc * 4 + khalf + j;
            const int d = c >> 3;
            const int hh = c & 7;
            afrag[kc][j] =
                acc[(size_t)gr * HDIM + hh * DHEAD + d] * inv[hh];
        }
    }

#pragma unroll
    for (int nt = 0; nt < 4; ++nt) {
        const int n0 = nt * 16;
        const float bb = bo[n0 + mrow];
        v8f c;
#pragma unroll
        for (int j = 0; j < 8; ++j) c[j] = bb;
#pragma unroll
        for (int kc = 0; kc < 16; ++kc) {
            const int kb = kc * 4 + khalf;
            v2f b;
            b.x = sW[(kb + 0) * HDIM + n0 + mrow];
            b.y = sW[(kb + 1) * HDIM + n0 + mrow];
            c = __builtin_amdgcn_wmma_f32_16x16x4_f32(
                    false, afrag[kc], false, b, (short)0, c, false, false);
        }
#pragma unroll
        for (int j = 0; j < 8; ++j) {
            const int m = half * 8 + j;
            const int grr = rowBase + m;
            if (grr < N) out[(size_t)grr * HDIM + n0 + mrow] = c[j];
        }
    }
}

// ---------------------------------------------------------------------------
// Host launcher
// ---------------------------------------------------------------------------
extern "C" void kernel_launch(void* const* d_in, const int* in_sizes, int n_in,
                              void* d_out, int out_size, void* d_ws, size_t ws_size,
                              hipStream_t stream) {
    const int*   X        = (const int*)d_in[0];
    const int*   row      = (const int*)d_in[1];
    const int*   col      = (const int*)d_in[2];
    const float* atom_emb = (const float*)d_in[3];
    const float* Wq = (const float*)d_in[4];
    const float* bq = (const float*)d_in[5];
    const float* Wk = (const float*)d_in[6];
    const float* bk = (const float*)d_in[7];
    const float* Wv = (const float*)d_in[8];
    const float* bv = (const float*)d_in[9];
    const float* Wo = (const float*)d_in[10];
    const float* bo = (const float*)d_in[11];
    float* out = (float*)d_out;

    const int N = in_sizes[0] / 9;
    const int E = in_sizes[1];

    // workspace carving (floats)
    float* ws = (float*)d_ws;
    const size_t nh = (size_t)N * HDIM;
    float*    h     = ws;            // reused as acc after QKV
    float*    q     = ws + nh;
    float*    k     = ws + 2 * nh;
    float*    v     = ws + 3 * nh;
    float*    score = ws + 4 * nh;   // E*8
    unsigned* m_ord = (unsigned*)(score + (size_t)E * NHEAD);
    float*    ssum  = (float*)(m_ord + (size_t)N * NHEAD);

    const int T = 256;
    const int gridNode = (N * HDIM + T - 1) / T;
    const int gridEdge = (E * NHEAD + T - 1) / T;
    const int gridGemm = (N + 127) / 128;

    atom_encoder_kernel<<<gridNode, T, 0, stream>>>(X, atom_emb, h, N);
    qkv_wmma_kernel<<<gridGemm, T, 0, stream>>>(h, Wq, bq, Wk, bk, Wv, bv,
                                                q, k, v, N);
    init_kernel<<<gridNode, T, 0, stream>>>(h /*acc*/, m_ord, ssum, N);
    sddmm_max_kernel<<<gridEdge, T, 0, stream>>>(q, k, row, col, score, m_ord, E);
    exp_spmm_kernel<<<gridEdge, T, 0, stream>>>(score, v, row, col, m_ord,
                                                ssum, h /*acc*/, E);
    out_proj_wmma_kernel<<<gridGemm, T, 0, stream>>>(h /*acc*/, ssum, Wo, bo,
                                                     out, N);
}